// DBLoss_42666205119058
// MI455X (gfx1250) — compile-verified
//
#include <hip/hip_runtime.h>
#include <math.h>
#include <stdint.h>

// ---------------- problem constants ----------------
#define NPIX      (640 * 640)      // 409600 pixels per sample
#define NSAMP     16
#define NPROB     32               // 16 samples x 2 maps (shrink, sigmoid(binary))
#define NB1       2048             // level-1 bins: float bits >> 19  (max 2032 < 2048)
#define NB2       1024             // level-2 bins: bits[18:9]
#define NB3       512              // level-3 bins: bits[8:0]
#define CHUNKS    16               // blocks per sample for streaming passes
#define CHUNK_PIX (NPIX / CHUNKS)  // 25600
#define TPB       256
#define ITERS     (CHUNK_PIX / TPB) // 100, uniform across the block (WMMA needs full EXEC)
#define EPSF      1e-7f
#define ONE_M_EPSF (1.0f - 1e-7f)
#define SENT      0xFFFFFFFFu

typedef __attribute__((ext_vector_type(2))) float v2f;
typedef __attribute__((ext_vector_type(8))) float v8f;

// ---------------- workspace layout ----------------
struct WSHead {
  float    pos_cnt[NSAMP];    // # positives (gt > 0.5) per sample
  float    thr_sel[NSAMP];    // threshold-loss selected count
  float    thr_l1[NSAMP];     // threshold-loss L1 sum
  float    pos_lsum[NPROB];   // sum of BCE over positives, per (sample,map)
  float    tot_lsum[NPROB];   // sum of BCE over all pixels, per (sample,map)
  unsigned b1[NPROB];         // level-1 winning bin (SENT => all-ones mask)
  unsigned krem1[NPROB];      // residual rank inside b1
  unsigned cnt_ab1[NPROB];    // # negatives strictly above bin b1
  float    lsum_ab1[NPROB];   // BCE sum of those
  unsigned b2[NPROB];
  unsigned krem2[NPROB];
  unsigned cnt_ab2[NPROB];    // cumulative count above (b1,b2)
  float    lsum_ab2[NPROB];
};
static_assert(sizeof(WSHead) <= 4096, "WSHead too big");

#define OFF_H1C 4096
#define SZ_H1   (NPROB * NB1 * 4)          // 256 KB
#define OFF_H1S (OFF_H1C + SZ_H1)
#define OFF_H2C (OFF_H1S + SZ_H1)
#define SZ_H2   (NPROB * NB2 * 4)          // 128 KB
#define OFF_H2S (OFF_H2C + SZ_H2)
#define OFF_H3C (OFF_H2S + SZ_H2)
#define SZ_H3   (NPROB * NB3 * 4)          // 64 KB
#define OFF_H3S (OFF_H3C + SZ_H3)
#define WS_TOTAL (OFF_H3S + SZ_H3)         // ~0.9 MB

// ---------------- async Global -> LDS staging (ASYNCcnt path) ----------------
// Builtin exists (probe-confirmed via previous round's diagnostic); clang wants
// int-typed pointers in AS1/AS3. Each lane reads back only its own slot, so no
// workgroup barrier is needed -- only s_wait_asynccnt.
__device__ __forceinline__ void async_load_f32(const float* gp, float* lp) {
#if defined(__gfx1250__) && __has_builtin(__builtin_amdgcn_global_load_async_to_lds_b32)
  __builtin_amdgcn_global_load_async_to_lds_b32(
      (__attribute__((address_space(1))) int*)(uintptr_t)gp,
      (__attribute__((address_space(3))) int*)(uint32_t)(uintptr_t)lp,
      0, 0);
#else
  *lp = *gp;
#endif
}

template <int N>
__device__ __forceinline__ void wait_async() {
#if defined(__gfx1250__) && __has_builtin(__builtin_amdgcn_s_wait_asynccnt)
  __builtin_amdgcn_s_wait_asynccnt(N);
#elif defined(__gfx1250__) && __has_builtin(__builtin_amdgcn_global_load_async_to_lds_b32)
  asm volatile("s_wait_asynccnt %0" :: "i"(N) : "memory");
#endif
}

// ---------------- wave32 sum via V_WMMA_F32_16X16X4_F32 ----------------
// A: a0 = v (A[m][0] lanes0-15, A[m][2] lanes16-31), a1 = 0. B = ones.
// D[m][n] = v(m) + v(m+16) -> per-lane sum of the 8 D vgprs gives half-sums,
// one shfl_xor(16) finishes. Exact for integer-valued inputs < 2^24.
__device__ __forceinline__ float wave_sum(float v) {
#if defined(__gfx1250__) && __has_builtin(__builtin_amdgcn_wmma_f32_16x16x4_f32)
  v2f a; a[0] = v;    a[1] = 0.0f;
  v2f b; b[0] = 1.0f; b[1] = 1.0f;
  v8f c = {0.f, 0.f, 0.f, 0.f, 0.f, 0.f, 0.f, 0.f};
  v8f d = __builtin_amdgcn_wmma_f32_16x16x4_f32(false, a, false, b, (short)0, c,
                                                false, false);
  float s = ((d[0] + d[1]) + (d[2] + d[3])) + ((d[4] + d[5]) + (d[6] + d[7]));
  s += __shfl_xor(s, 16, 32);
  return s;
#else
  for (int off = 16; off; off >>= 1) v += __shfl_xor(v, off, 32);
  return v;
#endif
}

// BCE value with the reference's clip. t is exactly 0 or 1 -> single log.
__device__ __forceinline__ float bce_l(float score, bool pos) {
  float pc = fminf(fmaxf(score, EPSF), ONE_M_EPSF);
  return pos ? -logf(pc) : -log1pf(-pc);
}

// ---------------- kernels ----------------
__global__ void k_zero(unsigned* ws, int nwords) {
  int i = blockIdx.x * blockDim.x + threadIdx.x;
  if (i < nwords) ws[i] = 0u;
}

// Pass 1 (the only HBM pass): stats + threshold loss + level-1 (cnt,lsum) hists.
// gt streams are staged through LDS with double-buffered async copies.
__global__ __launch_bounds__(TPB) void k_pass1(const float* __restrict__ outs,
                                               const float* __restrict__ gts,
                                               const float* __restrict__ gtt,
                                               char* __restrict__ ws) {
  __shared__ unsigned shc[2 * NB1];      // 16 KB
  __shared__ float    shs[2 * NB1];      // 16 KB
  __shared__ float    sh_t[2][TPB];      // 2 KB  (gt_shrink staging)
  __shared__ float    sh_g[2][TPB];      // 2 KB  (gt_threshold staging)
  const int s = blockIdx.y, chunk = blockIdx.x, tid = threadIdx.x;
  const float* shr = outs + (size_t)s * 3 * NPIX;
  const float* thm = shr + NPIX;
  const float* bnm = shr + 2 * NPIX;
  const float* g0  = gts + (size_t)s * NPIX;
  const float* g1  = gtt + (size_t)s * NPIX;
  for (int i = tid; i < 2 * NB1; i += TPB) { shc[i] = 0u; shs[i] = 0.0f; }
  __syncthreads();

  float a_pos = 0, a_sel = 0, a_l1 = 0, a_pl0 = 0, a_tl0 = 0, a_pl1 = 0, a_tl1 = 0;
  const int base = chunk * CHUNK_PIX + tid;
  async_load_f32(&g0[base], &sh_t[0][tid]);
  async_load_f32(&g1[base], &sh_g[0][tid]);
  for (int it = 0; it < ITERS; ++it) {
    int p = base + it * TPB;
    int cur = it & 1;
    if (it + 1 < ITERS) {
      async_load_f32(&g0[p + TPB], &sh_t[cur ^ 1][tid]);
      async_load_f32(&g1[p + TPB], &sh_g[cur ^ 1][tid]);
      __builtin_prefetch(&shr[p + TPB], 0, 0);   // global_prefetch_b8
      __builtin_prefetch(&bnm[p + TPB], 0, 0);
      wait_async<2>();                           // current pair complete
    } else {
      wait_async<0>();
    }
    float v0 = shr[p], v1 = thm[p], v2 = bnm[p];
    float t = sh_t[cur][tid], gt = sh_g[cur][tid];

    bool pos = t > 0.5f;
    bool sel = (gt > 0.0f) || (t > 0.0f);
    a_sel += sel ? 1.0f : 0.0f;
    a_l1  += sel ? fabsf(v1 - gt) : 0.0f;
    a_pos += pos ? 1.0f : 0.0f;

    float l0 = bce_l(v0, pos);
    a_tl0 += l0;
    unsigned bits0 = __float_as_uint(v0);
    float sg = 1.0f / (1.0f + expf(-v2));
    float l1v = bce_l(sg, pos);
    a_tl1 += l1v;
    unsigned bits1 = __float_as_uint(sg);

    if (pos) { a_pl0 += l0; a_pl1 += l1v; }
    else {
      atomicAdd(&shc[bits0 >> 19], 1u);
      atomicAdd(&shs[bits0 >> 19], l0);
      atomicAdd(&shc[NB1 + (bits1 >> 19)], 1u);
      atomicAdd(&shs[NB1 + (bits1 >> 19)], l1v);
    }
  }
  __syncthreads();

  unsigned* h1c = (unsigned*)(ws + OFF_H1C);
  float*    h1s = (float*)(ws + OFF_H1S);
  for (int i = tid; i < 2 * NB1; i += TPB) {
    int m = i >= NB1, b = i & (NB1 - 1);
    int idx = (s * 2 + m) * NB1 + b;
    unsigned c = shc[i]; float sv = shs[i];
    if (c)          atomicAdd(&h1c[idx], c);
    if (sv != 0.f)  atomicAdd(&h1s[idx], sv);
  }

  // scalar reductions: full EXEC here (uniform control flow), WMMA-legal
  WSHead* W = (WSHead*)ws;
  int lane = tid & 31;
  float r;
  r = wave_sum(a_pos); if (!lane) atomicAdd(&W->pos_cnt[s], r);
  r = wave_sum(a_sel); if (!lane) atomicAdd(&W->thr_sel[s], r);
  r = wave_sum(a_l1);  if (!lane) atomicAdd(&W->thr_l1[s], r);
  r = wave_sum(a_pl0); if (!lane) atomicAdd(&W->pos_lsum[s * 2 + 0], r);
  r = wave_sum(a_tl0); if (!lane) atomicAdd(&W->tot_lsum[s * 2 + 0], r);
  r = wave_sum(a_pl1); if (!lane) atomicAdd(&W->pos_lsum[s * 2 + 1], r);
  r = wave_sum(a_tl1); if (!lane) atomicAdd(&W->tot_lsum[s * 2 + 1], r);
}

// One warp per problem: scan level-1 histogram from the top, find bin of the
// k-th largest negative score (k = min(3*pos, neg)).
__global__ __launch_bounds__(1024) void k_find1(char* __restrict__ ws) {
  WSHead* W = (WSHead*)ws;
  const unsigned* h1c = (const unsigned*)(ws + OFF_H1C);
  const float*    h1s = (const float*)(ws + OFF_H1S);
  int p = threadIdx.x >> 5, lane = threadIdx.x & 31, s = p >> 1;
  unsigned posi = (unsigned)(W->pos_cnt[s] + 0.5f);
  unsigned negc = NPIX - posi;
  unsigned k = (3u * posi < negc) ? 3u * posi : negc;
  if (posi == 0u || k == 0u) {  // all-ones mask path
    if (!lane) { W->b1[p] = SENT; W->krem1[p] = 0; W->cnt_ab1[p] = 0; W->lsum_ab1[p] = 0.f; }
    return;
  }
  unsigned cum = 0, bfound = 0, krem = 0; float ls = 0.f; bool found = false;
  for (int b0 = NB1 - 32; b0 >= 0 && !found; b0 -= 32) {
    float cf = (float)h1c[p * NB1 + b0 + lane];
    float sf = h1s[p * NB1 + b0 + lane];
    float ct = wave_sum(cf);           // exact: integer-valued, < 2^24
    float st = wave_sum(sf);
    unsigned cti = (unsigned)(ct + 0.5f);
    if (cum + cti < k) { cum += cti; ls += st; continue; }
    for (int j = 31; j >= 0; --j) {    // walk the found chunk top-down
      unsigned cj = (unsigned)(__shfl(cf, j, 32) + 0.5f);
      float    sj = __shfl(sf, j, 32);
      if (cum + cj >= k) { bfound = (unsigned)(b0 + j); krem = k - cum; found = true; break; }
      cum += cj; ls += sj;
    }
  }
  if (!lane) { W->b1[p] = bfound; W->krem1[p] = krem; W->cnt_ab1[p] = cum; W->lsum_ab1[p] = ls; }
}

// Pass 2 (L2-resident): level-2 (cnt,lsum) hist for negatives landing in b1.
__global__ __launch_bounds__(TPB) void k_pass2(const float* __restrict__ outs,
                                               const float* __restrict__ gts,
                                               char* __restrict__ ws) {
  __shared__ unsigned shc[2 * NB2];
  __shared__ float    shs[2 * NB2];
  __shared__ float    sh_t[2][TPB];
  const int s = blockIdx.y, chunk = blockIdx.x, tid = threadIdx.x;
  WSHead* W = (WSHead*)ws;
  const unsigned t0 = W->b1[s * 2 + 0], t1 = W->b1[s * 2 + 1];
  const float* shr = outs + (size_t)s * 3 * NPIX;
  const float* bnm = shr + 2 * NPIX;
  const float* g0  = gts + (size_t)s * NPIX;
  for (int i = tid; i < 2 * NB2; i += TPB) { shc[i] = 0u; shs[i] = 0.0f; }
  __syncthreads();
  const int base = chunk * CHUNK_PIX + tid;
  async_load_f32(&g0[base], &sh_t[0][tid]);
  for (int it = 0; it < ITERS; ++it) {
    int p = base + it * TPB;
    int cur = it & 1;
    if (it + 1 < ITERS) {
      async_load_f32(&g0[p + TPB], &sh_t[cur ^ 1][tid]);
      wait_async<1>();
    } else {
      wait_async<0>();
    }
    float v0 = shr[p], v2 = bnm[p], t = sh_t[cur][tid];
    if (t <= 0.5f) {
      unsigned bits0 = __float_as_uint(v0);
      if ((bits0 >> 19) == t0) {
        float l = bce_l(v0, false); unsigned b = (bits0 >> 9) & (NB2 - 1);
        atomicAdd(&shc[b], 1u); atomicAdd(&shs[b], l);
      }
      float sg = 1.0f / (1.0f + expf(-v2));
      unsigned bits1 = __float_as_uint(sg);
      if ((bits1 >> 19) == t1) {
        float l = bce_l(sg, false); unsigned b = (bits1 >> 9) & (NB2 - 1);
        atomicAdd(&shc[NB2 + b], 1u); atomicAdd(&shs[NB2 + b], l);
      }
    }
  }
  __syncthreads();
  unsigned* h2c = (unsigned*)(ws + OFF_H2C);
  float*    h2s = (float*)(ws + OFF_H2S);
  for (int i = tid; i < 2 * NB2; i += TPB) {
    int m = i >= NB2, b = i & (NB2 - 1);
    int idx = (s * 2 + m) * NB2 + b;
    unsigned c = shc[i]; float sv = shs[i];
    if (c)         atomicAdd(&h2c[idx], c);
    if (sv != 0.f) atomicAdd(&h2s[idx], sv);
  }
}

__global__ __launch_bounds__(1024) void k_find2(char* __restrict__ ws) {
  WSHead* W = (WSHead*)ws;
  const unsigned* h2c = (const unsigned*)(ws + OFF_H2C);
  const float*    h2s = (const float*)(ws + OFF_H2S);
  int p = threadIdx.x >> 5, lane = threadIdx.x & 31;
  if (W->b1[p] == SENT) {
    if (!lane) { W->b2[p] = SENT; W->krem2[p] = 0; W->cnt_ab2[p] = 0; W->lsum_ab2[p] = 0.f; }
    return;
  }
  unsigned k = W->krem1[p];
  unsigned cum = 0, bfound = 0, krem = 0; float ls = 0.f; bool found = false;
  for (int b0 = NB2 - 32; b0 >= 0 && !found; b0 -= 32) {
    float cf = (float)h2c[p * NB2 + b0 + lane];
    float sf = h2s[p * NB2 + b0 + lane];
    float ct = wave_sum(cf);
    float st = wave_sum(sf);
    unsigned cti = (unsigned)(ct + 0.5f);
    if (cum + cti < k) { cum += cti; ls += st; continue; }
    for (int j = 31; j >= 0; --j) {
      unsigned cj = (unsigned)(__shfl(cf, j, 32) + 0.5f);
      float    sj = __shfl(sf, j, 32);
      if (cum + cj >= k) { bfound = (unsigned)(b0 + j); krem = k - cum; found = true; break; }
      cum += cj; ls += sj;
    }
  }
  if (!lane) {
    W->b2[p] = bfound; W->krem2[p] = krem;
    W->cnt_ab2[p]  = W->cnt_ab1[p] + cum;
    W->lsum_ab2[p] = W->lsum_ab1[p] + ls;
  }
}

// Pass 3 (L2-resident): level-3 hist for negatives matching (b1,b2).
__global__ __launch_bounds__(TPB) void k_pass3(const float* __restrict__ outs,
                                               const float* __restrict__ gts,
                                               char* __restrict__ ws) {
  __shared__ unsigned shc[2 * NB3];
  __shared__ float    shs[2 * NB3];
  __shared__ float    sh_t[2][TPB];
  const int s = blockIdx.y, chunk = blockIdx.x, tid = threadIdx.x;
  WSHead* W = (WSHead*)ws;
  const unsigned t0 = W->b1[s * 2 + 0], t1 = W->b1[s * 2 + 1];
  const unsigned u0 = W->b2[s * 2 + 0], u1 = W->b2[s * 2 + 1];
  const float* shr = outs + (size_t)s * 3 * NPIX;
  const float* bnm = shr + 2 * NPIX;
  const float* g0  = gts + (size_t)s * NPIX;
  for (int i = tid; i < 2 * NB3; i += TPB) { shc[i] = 0u; shs[i] = 0.0f; }
  __syncthreads();
  const int base = chunk * CHUNK_PIX + tid;
  async_load_f32(&g0[base], &sh_t[0][tid]);
  for (int it = 0; it < ITERS; ++it) {
    int p = base + it * TPB;
    int cur = it & 1;
    if (it + 1 < ITERS) {
      async_load_f32(&g0[p + TPB], &sh_t[cur ^ 1][tid]);
      wait_async<1>();
    } else {
      wait_async<0>();
    }
    float v0 = shr[p], v2 = bnm[p], t = sh_t[cur][tid];
    if (t <= 0.5f) {
      unsigned bits0 = __float_as_uint(v0);
      if ((bits0 >> 19) == t0 && ((bits0 >> 9) & (NB2 - 1)) == u0) {
        float l = bce_l(v0, false); unsigned b = bits0 & (NB3 - 1);
        atomicAdd(&shc[b], 1u); atomicAdd(&shs[b], l);
      }
      float sg = 1.0f / (1.0f + expf(-v2));
      unsigned bits1 = __float_as_uint(sg);
      if ((bits1 >> 19) == t1 && ((bits1 >> 9) & (NB2 - 1)) == u1) {
        float l = bce_l(sg, false); unsigned b = bits1 & (NB3 - 1);
        atomicAdd(&shc[NB3 + b], 1u); atomicAdd(&shs[NB3 + b], l);
      }
    }
  }
  __syncthreads();
  unsigned* h3c = (unsigned*)(ws + OFF_H3C);
  float*    h3s = (float*)(ws + OFF_H3S);
  for (int i = tid; i < 2 * NB3; i += TPB) {
    int m = i >= NB3, b = i & (NB3 - 1);
    int idx = (s * 2 + m) * NB3 + b;
    unsigned c = shc[i]; float sv = shs[i];
    if (c)         atomicAdd(&h3c[idx], c);
    if (sv != 0.f) atomicAdd(&h3s[idx], sv);
  }
}

// Finalize: exact threshold suffix sums -> per-sample losses -> 4 outputs.
__global__ void k_final(const char* __restrict__ ws, float* __restrict__ out) {
  const WSHead* W = (const WSHead*)ws;
  const unsigned* h3c = (const unsigned*)(ws + OFF_H3C);
  const float*    h3s = (const float*)(ws + OFF_H3S);
  __shared__ float lossm[NPROB];
  __shared__ float lt[NSAMP];
  int tid = threadIdx.x;
  if (tid < NPROB) {
    int p = tid, s = p >> 1;
    float loss;
    if (W->b1[p] == SENT) {
      loss = W->tot_lsum[p] / (float)NPIX;          // all-ones mask
    } else {
      unsigned k = W->krem2[p], cum = 0; float ls = 0.f;
      unsigned selC = 0; float selL = 0.f;
      for (int b = NB3 - 1; b >= 0; --b) {
        unsigned c = h3c[p * NB3 + b]; float sv = h3s[p * NB3 + b];
        if (cum + c >= k) {                         // thr bin found; >= thr includes ties
          selC = W->cnt_ab2[p] + cum + c;
          selL = W->lsum_ab2[p] + ls + sv;
          break;
        }
        cum += c; ls += sv;
      }
      float denom = W->pos_cnt[s] + (float)selC;
      loss = (W->pos_lsum[p] + selL) / fmaxf(denom, 1.0f);
    }
    lossm[p] = loss;
  }
  if (tid < NSAMP) {
    float sc = W->thr_sel[tid];
    lt[tid] = sc > 0.f ? (W->thr_l1[tid] / fmaxf(sc, 1.0f)) : 0.f;
  }
  __syncthreads();
  if (tid == 0) {
    float a = 0.f, b = 0.f, c = 0.f;
    for (int s2 = 0; s2 < NSAMP; ++s2) {
      a += lossm[2 * s2]; b += lossm[2 * s2 + 1]; c += lt[s2];
    }
    a *= (1.0f / NSAMP); b *= (1.0f / NSAMP); c *= (1.0f / NSAMP);
    out[0] = a + 1.0f * b + 10.0f * c;  // loss_all (ALPHA=1, BETA=10)
    out[1] = a;                         // loss_shrink
    out[2] = b;                         // loss_binary
    out[3] = c;                         // loss_threshold
  }
}

extern "C" void kernel_launch(void* const* d_in, const int* in_sizes, int n_in,
                              void* d_out, int out_size, void* d_ws, size_t ws_size,
                              hipStream_t stream) {
  (void)in_sizes; (void)n_in; (void)out_size; (void)ws_size;
  const float* outs = (const float*)d_in[0];   // [16,3,640,640]
  const float* gts  = (const float*)d_in[1];   // [16,640,640]
  const float* gtt  = (const float*)d_in[2];   // [16,640,640]
  char* ws = (char*)d_ws;

  int nwords = WS_TOTAL / 4;
  k_zero<<<(nwords + 255) / 256, 256, 0, stream>>>((unsigned*)ws, nwords);
  k_pass1<<<dim3(CHUNKS, NSAMP), TPB, 0, stream>>>(outs, gts, gtt, ws);
  k_find1<<<1, 1024, 0, stream>>>(ws);
  k_pass2<<<dim3(CHUNKS, NSAMP), TPB, 0, stream>>>(outs, gts, ws);
  k_find2<<<1, 1024, 0, stream>>>(ws);
  k_pass3<<<dim3(CHUNKS, NSAMP), TPB, 0, stream>>>(outs, gts, ws);
  k_final<<<1, 64, 0, stream>>>(ws, (float*)d_out);
}